// PPDFN_88656714925429
// MI455X (gfx1250) — compile-verified
//
#include <hip/hip_runtime.h>
#include <stdint.h>

typedef __attribute__((ext_vector_type(2))) float v2f;
typedef __attribute__((ext_vector_type(8))) float v8f;

#define C_   32
#define H_   128
#define W_   128
#define FS   5
#define PAD  2
#define TW   16            // tile width  (pixels)
#define TH   8             // tile height (pixels) == waves per block
#define HWT  (TW + 2*PAD)  // 20 (tile cols + halo)
#define HHT  (TH + 2*PAD)  // 12 (tile rows + halo)
#define CP   33            // padded channel stride in LDS (odd -> conflict-free)
#define SFS  27            // padded filter-row stride (odd -> conflict-free)

__global__ __launch_bounds__(256)
void ppdfn_dynfilter_kernel(const float* __restrict__ x,
                            const float* __restrict__ wg,
                            const float* __restrict__ bg,
                            float* __restrict__ out)
{
    __shared__ float sX[HHT * HWT * CP];   // [row][col][channel], halo tile of x
    __shared__ float sW[32 * 32];          // w_gen padded to 32 filters (rows 25..31 = 0)
    __shared__ float sB[32];               // bias padded
    __shared__ float sF[TH * TW * SFS];    // per-pixel dynamic filters [pixel][tap]

    const int tid  = threadIdx.x;
    const int lane = tid & 31;
    const int wv   = tid >> 5;             // wave id 0..7 == tile row

    const int tx0 = blockIdx.x * TW;
    const int ty0 = blockIdx.y * TH;
    const int b   = blockIdx.z;

    // ---------- Stage 0: global -> LDS halo tile via CDNA5 async-to-LDS ----------
    // GLOBAL_LOAD_ASYNC_TO_LDS_B32 (GVS: SGPR64 base + VGPR32 offset) writes LDS
    // directly, no VGPR round-trip; tracked by ASYNCcnt. OOB halo lanes store 0.
    const float* xb = x + (size_t)b * C_ * H_ * W_;
    for (int e = tid; e < HHT * HWT * C_; e += 256) {
        const int col = e % HWT;                 // fastest: coalesced-ish in w
        const int row = (e / HWT) % HHT;
        const int c   = e / (HWT * HHT);
        const int gh  = ty0 + row - PAD;
        const int gw  = tx0 + col - PAD;
        float* sp = &sX[(row * HWT + col) * CP + c];
        if (gh >= 0 && gh < H_ && gw >= 0 && gw < W_) {
            const unsigned ldsAddr = (unsigned)(uintptr_t)sp;        // addr[31:0] = LDS offset
            const unsigned gOff    = (unsigned)(((c * H_ + gh) * W_ + gw) * 4);
            asm volatile("global_load_async_to_lds_b32 %0, %1, %2"
                         :: "v"(ldsAddr), "v"(gOff), "s"(xb) : "memory");
        } else {
            *sp = 0.0f;                                              // zero-fill halo OOB
        }
    }
    for (int e = tid; e < 32 * 32; e += 256) {
        const int f = e >> 5, c = e & 31;
        sW[e] = (f < FS * FS) ? wg[f * C_ + c] : 0.0f;
    }
    if (tid < 32) sB[tid] = (tid < FS * FS) ? bg[tid] : 0.0f;
    asm volatile("s_wait_asynccnt 0x0" ::: "memory");                // async fills done
    __syncthreads();

    // ---------- Stage 1: per-pixel filters via V_WMMA_F32_16X16X4_F32 ----------
    // M = 16 pixels (this wave's tile row), N = 16 filters x 2 tiles, K = 32 (8 steps of 4)
    const int m    = lane & 15;            // pixel column / filter index within N-tile
    const int half = lane >> 4;            // K-pair select per ISA A/B layouts

    const float* aRow  = &sX[((wv + PAD) * HWT + (m + PAD)) * CP]; // x[pixel][c..]
    const float* bRow0 = &sW[m * 32];                              // w_gen[f=m][c..]
    const float* bRow1 = &sW[(m + 16) * 32];                       // w_gen[f=m+16][c..]

    v8f acc0 = {};
    v8f acc1 = {};
    #pragma unroll
    for (int kk = 0; kk < 8; ++kk) {
        const int kb = kk * 4 + half * 2;  // lanes 0-15: K=k0,k0+1 ; lanes 16-31: K=k0+2,k0+3
        v2f a;  a.x  = aRow[kb];   a.y  = aRow[kb + 1];
        v2f b0; b0.x = bRow0[kb];  b0.y = bRow0[kb + 1];
        v2f b1; b1.x = bRow1[kb];  b1.y = bRow1[kb + 1];
        acc0 = __builtin_amdgcn_wmma_f32_16x16x4_f32(false, a, false, b0,
                                                     (short)0, acc0, false, false);
        acc1 = __builtin_amdgcn_wmma_f32_16x16x4_f32(false, a, false, b1,
                                                     (short)0, acc1, false, false);
    }

    // Scatter filters (+bias) to LDS: C/D layout vgpr r, lane l -> M=r+8*(l>>4), N=l&15
    {
        const int prow = wv * TW;
        #pragma unroll
        for (int r = 0; r < 8; ++r) {
            const int p = prow + r + half * 8;
            sF[p * SFS + m]      = acc0[r] + sB[m];
            sF[p * SFS + m + 16] = acc1[r] + sB[m + 16];
        }
    }
    // Each wave reads back only its own sF rows -> in-wave DS ordering is enough,
    // no workgroup barrier required here.

    // ---------- Stage 2: apply dynamic 5x5 filter, 16 channels per lane ----------
    {
        const float* fRow = &sF[(wv * TW + m) * SFS];
        float acc[16];
        #pragma unroll
        for (int cc = 0; cc < 16; ++cc) acc[cc] = 0.0f;

        #pragma unroll
        for (int dy = 0; dy < FS; ++dy) {
            #pragma unroll
            for (int dx = 0; dx < FS; ++dx) {
                const float fv = fRow[dy * FS + dx];   // tap order matches reference
                const float* xp = &sX[((wv + dy) * HWT + (m + dx)) * CP + half * 16];
                #pragma unroll
                for (int cc = 0; cc < 16; ++cc)
                    acc[cc] = fmaf(xp[cc], fv, acc[cc]);
            }
        }

        const int gh = ty0 + wv;
        float* ob = out + ((size_t)b * C_ + half * 16) * (H_ * W_)
                        + gh * W_ + tx0 + m;
        #pragma unroll
        for (int cc = 0; cc < 16; ++cc)
            ob[(size_t)cc * (H_ * W_)] = acc[cc];      // 16 consecutive floats/half-wave
    }
}

extern "C" void kernel_launch(void* const* d_in, const int* in_sizes, int n_in,
                              void* d_out, int out_size, void* d_ws, size_t ws_size,
                              hipStream_t stream) {
    (void)n_in; (void)out_size; (void)d_ws; (void)ws_size;
    const float* x  = (const float*)d_in[0];
    const float* wg = (const float*)d_in[1];
    const float* bg = (const float*)d_in[2];
    float* out = (float*)d_out;

    const int B = in_sizes[0] / (C_ * H_ * W_);   // 8
    dim3 grid(W_ / TW, H_ / TH, B);               // (8, 16, 8)
    ppdfn_dynfilter_kernel<<<grid, 256, 0, stream>>>(x, wg, bg, out);
}